// Net_85048942396282
// MI455X (gfx1250) — compile-verified
//
#include <hip/hip_runtime.h>
#include <math.h>

// ---------------- complex helpers ----------------
struct cplx { float x, y; };
__device__ __forceinline__ cplx cmk(float a, float b){ cplx r; r.x=a; r.y=b; return r; }
__device__ __forceinline__ cplx cadd(cplx a, cplx b){ return cmk(a.x+b.x, a.y+b.y); }
__device__ __forceinline__ cplx csub(cplx a, cplx b){ return cmk(a.x-b.x, a.y-b.y); }
__device__ __forceinline__ cplx cscale(cplx a, float s){ return cmk(a.x*s, a.y*s); }
__device__ __forceinline__ cplx cmul(cplx a, cplx b){ return cmk(a.x*b.x - a.y*b.y, a.x*b.y + a.y*b.x); }

// ---------------- 4-qubit (per-lane, 16 amps in VGPRs) ----------------
// flat idx: qubit0 = bit3 ... qubit3 = bit0 ; mask(q) = 8>>q
__device__ __forceinline__ void h4(cplx* st, int m){
  const float r2 = 0.70710678118654752f;
  #pragma unroll
  for (int i=0;i<16;i++) if (!(i&m)){ int j=i|m; cplx a=st[i], b=st[j];
    st[i]=cscale(cadd(a,b),r2); st[j]=cscale(csub(a,b),r2); }
}
__device__ __forceinline__ void rz4(cplx* st, int m, float th){
  float s,c; __sincosf(0.5f*th,&s,&c);
  #pragma unroll
  for (int i=0;i<16;i++){ cplx ph = cmk(c, (i&m)? s : -s); st[i]=cmul(st[i],ph); }
}
__device__ __forceinline__ void rx4(cplx* st, int m, float th){
  float s,c; __sincosf(0.5f*th,&s,&c);
  #pragma unroll
  for (int i=0;i<16;i++) if (!(i&m)){ int j=i|m; cplx a=st[i], b=st[j];
    st[i]=cmk(c*a.x + s*b.y, c*a.y - s*b.x);
    st[j]=cmk(s*a.y + c*b.x, -s*a.x + c*b.y); }
}
__device__ __forceinline__ void ry4(cplx* st, int m, float th){
  float s,c; __sincosf(0.5f*th,&s,&c);
  #pragma unroll
  for (int i=0;i<16;i++) if (!(i&m)){ int j=i|m; cplx a=st[i], b=st[j];
    st[i]=cmk(c*a.x - s*b.x, c*a.y - s*b.y);
    st[j]=cmk(s*a.x + c*b.x, s*a.y + c*b.y); }
}
__device__ __forceinline__ void crz4(cplx* st, int mc, int mt, float th){
  float s,c; __sincosf(0.5f*th,&s,&c);
  #pragma unroll
  for (int i=0;i<16;i++){
    cplx ph = (i&mc) ? cmk(c, (i&mt)? s : -s) : cmk(1.f, 0.f);
    st[i]=cmul(st[i],ph);
  }
}

// ---------------- 8-qubit (16-lane half-wave cooperative, 16 amps/lane) ----
// flat idx (8 bits): qubit0=bit7 ... qubit7=bit0. lane-in-half lh = bits 7:4,
// register r = bits 3:0. Qubits 0..3 -> shfl_xor lanes; qubits 4..7 -> regs.
__device__ __forceinline__ void h8(cplx* st, int lh, int q){
  const float r2 = 0.70710678118654752f;
  if (q >= 4){
    const int m = 8 >> (q-4);
    #pragma unroll
    for (int i=0;i<16;i++) if (!(i&m)){ int j=i|m; cplx a=st[i], b=st[j];
      st[i]=cscale(cadd(a,b),r2); st[j]=cscale(csub(a,b),r2); }
  } else {
    const int lm = 8 >> q;
    const bool hi = (lh & lm)!=0;
    #pragma unroll
    for (int r=0;r<16;r++){
      cplx o=st[r];
      cplx p = cmk(__shfl_xor(o.x, lm, 32), __shfl_xor(o.y, lm, 32));
      st[r] = hi ? cscale(csub(p,o),r2) : cscale(cadd(o,p),r2);
    }
  }
}
__device__ __forceinline__ void rx8(cplx* st, int lh, int q, float th){
  float s,c; __sincosf(0.5f*th,&s,&c);
  if (q >= 4){
    const int m = 8 >> (q-4);
    #pragma unroll
    for (int i=0;i<16;i++) if (!(i&m)){ int j=i|m; cplx a=st[i], b=st[j];
      st[i]=cmk(c*a.x + s*b.y, c*a.y - s*b.x);
      st[j]=cmk(s*a.y + c*b.x, -s*a.x + c*b.y); }
  } else {
    const int lm = 8 >> q;
    #pragma unroll
    for (int r=0;r<16;r++){
      cplx o=st[r];
      cplx p = cmk(__shfl_xor(o.x, lm, 32), __shfl_xor(o.y, lm, 32));
      // new = c*own + (-i s)*recv  (RX matrix is symmetric)
      st[r] = cmk(c*o.x + s*p.y, c*o.y - s*p.x);
    }
  }
}
__device__ __forceinline__ void ry8(cplx* st, int lh, int q, float th){
  float s,c; __sincosf(0.5f*th,&s,&c);
  if (q >= 4){
    const int m = 8 >> (q-4);
    #pragma unroll
    for (int i=0;i<16;i++) if (!(i&m)){ int j=i|m; cplx a=st[i], b=st[j];
      st[i]=cmk(c*a.x - s*b.x, c*a.y - s*b.y);
      st[j]=cmk(s*a.x + c*b.x, s*a.y + c*b.y); }
  } else {
    const int lm = 8 >> q;
    const bool hi = (lh & lm)!=0;
    const float ss = hi ? s : -s;
    #pragma unroll
    for (int r=0;r<16;r++){
      cplx o=st[r];
      cplx p = cmk(__shfl_xor(o.x, lm, 32), __shfl_xor(o.y, lm, 32));
      st[r] = cmk(c*o.x + ss*p.x, c*o.y + ss*p.y);
    }
  }
}
__device__ __forceinline__ void rz8(cplx* st, int lh, int q, float th){
  float s,c; __sincosf(0.5f*th,&s,&c);
  #pragma unroll
  for (int r=0;r<16;r++){
    int idx=(lh<<4)|r; int b=(idx>>(7-q))&1;
    cplx ph = cmk(c, b? s : -s); st[r]=cmul(st[r],ph);
  }
}
__device__ __forceinline__ void crz8(cplx* st, int lh, int cw, int tw, float th){
  float s,c; __sincosf(0.5f*th,&s,&c);
  #pragma unroll
  for (int r=0;r<16;r++){
    int idx=(lh<<4)|r;
    int cb=(idx>>(7-cw))&1, tb=(idx>>(7-tw))&1;
    cplx ph = cb ? cmk(c, tb? s : -s) : cmk(1.f, 0.f);
    st[r]=cmul(st[r],ph);
  }
}

// ---------------- kernel 1: quanvolution + circuit8 ----------------
// one block = 16 samples; one wave = 2 samples (16-lane halves)
__global__ __launch_bounds__(256) void qnn_main(const float* __restrict__ X,
    const float* __restrict__ w1, const float* __restrict__ w2,
    float* __restrict__ g, int bs) {
  __shared__ float Xs[16*196];
  __shared__ float fs[16*64];
  const int tid = threadIdx.x;
  const int s0 = blockIdx.x * 16;
  if (s0 >= bs) return;
  const int nstage = (bs - s0 >= 16 ? 16 : bs - s0) * 196;
  for (int k = tid; k < nstage; k += 256) Xs[k] = X[s0*196 + k];
  __syncthreads();

  const int lane = tid & 31;
  const int wv   = tid >> 5;
  const int h    = (lane >> 4) & 1;
  const int lh   = lane & 15;
  const int sIB  = wv*2 + h;                 // sample-in-block 0..15
  const float* xs = &Xs[sIB*196];

  // ---- phase A: circuit4, one 4x4 patch per lane ----
  {
    const int p  = lh;
    const int pi = p >> 2, pj = p & 3;
    const int i0 = pi*4, j0 = pj*4;
    const int cols = (j0==12)?2:4;
    const int rows = (i0==12)?2:4;
    const int L = rows*cols;
    cplx st[16];
    #pragma unroll
    for (int i=0;i<16;i++) st[i]=cmk(0.f,0.f);
    st[0]=cmk(1.f,0.f);
    #pragma unroll
    for (int q=0;q<4;q++){
      h4(st, 8>>q);
      #pragma unroll
      for (int ii=0;ii<4;ii++){
        const int idx = q*4+ii;
        if (idx < L){
          const int rr = (cols==4)? (idx>>2) : (idx>>1);
          const int cc = (cols==4)? (idx&3)  : (idx&1);
          const float th = xs[(i0+rr)*14 + (j0+cc)];
          if ((ii&1)==0) rz4(st, 8>>q, th); else rx4(st, 8>>q, th);
        }
      }
    }
    #pragma unroll
    for (int l=0;l<2;l++){
      #pragma unroll
      for (int i=0;i<4;i++) crz4(st, 8>>i, 8>>((i+1)&3), w1[l*8+i]);
      #pragma unroll
      for (int j=0;j<4;j++) ry4(st, 8>>j, w1[l*8+4+j]);
    }
    float pr[16];
    #pragma unroll
    for (int i=0;i<16;i++) pr[i]=st[i].x*st[i].x + st[i].y*st[i].y;
    #pragma unroll
    for (int w=0;w<4;w++){
      const int m = 8>>w; float sum=0.f;
      #pragma unroll
      for (int i=0;i<16;i++) sum += (i&m)? -pr[i] : pr[i];
      fs[sIB*64 + p*4 + w] = sum;
    }
  }
  __syncthreads();

  // ---- phase B: circuit8, 16 lanes cooperate on 256 amps ----
  const float* f = &fs[sIB*64];
  cplx s8[16];
  #pragma unroll
  for (int r=0;r<16;r++) s8[r]=cmk(0.f,0.f);
  if (lh==0) s8[0]=cmk(1.f,0.f);
  const float PI_ = 3.14159265358979323846f;
  #pragma unroll
  for (int q=0;q<8;q++){
    h8(s8, lh, q);
    #pragma unroll
    for (int ii=0;ii<9;ii++){
      const int idx = q*9+ii;
      if (idx < 64){
        const float th = PI_ * f[idx];
        if ((ii&1)==0) rz8(s8,lh,q,th); else rx8(s8,lh,q,th);
      }
    }
  }
  #pragma unroll
  for (int l=0;l<3;l++){
    #pragma unroll
    for (int i=0;i<8;i++) crz8(s8, lh, i, (i+1)&7, w2[l*16+i]);
    #pragma unroll
    for (int j=0;j<8;j++) ry8(s8, lh, j, w2[l*16+8+j]);
  }
  // Z expectations
  float pr[16];
  #pragma unroll
  for (int r=0;r<16;r++) pr[r]=s8[r].x*s8[r].x + s8[r].y*s8[r].y;
  float tot=0.f;
  #pragma unroll
  for (int r=0;r<16;r++) tot += pr[r];
  float part[8];
  #pragma unroll
  for (int w=0;w<4;w++) part[w] = (lh & (8>>w)) ? -tot : tot;
  #pragma unroll
  for (int w=4;w<8;w++){
    const int m = 8>>(w-4); float sum=0.f;
    #pragma unroll
    for (int r=0;r<16;r++) sum += (r&m)? -pr[r] : pr[r];
    part[w]=sum;
  }
  #pragma unroll
  for (int w=0;w<8;w++){
    #pragma unroll
    for (int d=1; d<16; d<<=1) part[w] += __shfl_xor(part[w], d, 32);
  }
  const int sample = s0 + sIB;
  if (lh==0 && sample < bs){
    #pragma unroll
    for (int w=0;w<8;w++) g[sample*8 + w] = part[w];
  }
}

// ---------------- kernel 2: FC head via V_WMMA_F32_16X16X4_F32 ------------
typedef float v2f __attribute__((ext_vector_type(2)));
typedef float v8f __attribute__((ext_vector_type(8)));

__global__ __launch_bounds__(256) void fc_wmma_kernel(const float* __restrict__ g,
    const float* __restrict__ fc_w, const float* __restrict__ fc_b,
    float* __restrict__ out, int bs) {
  const int lane = threadIdx.x & 31;
  const int tile = blockIdx.x * (blockDim.x >> 5) + (threadIdx.x >> 5); // 16 samples/tile
  if (tile * 16 >= bs) return;          // wave-uniform guard: EXEC stays all-ones
  const int half = lane >> 4;           // 0: K={0,1}, M=0..7 ; 1: K={2,3}, M=8..15
  const int sl   = lane & 15;
  const int kh   = half * 2;
  // A tile: 16 samples x K   (lane = M row, vgpr = K per ISA 16x4 f32 layout)
  const float* ga = g + (tile*16 + sl)*8;
  v2f a0; a0.x = ga[kh];     a0.y = ga[kh+1];
  v2f a1; a1.x = ga[4+kh];   a1.y = ga[4+kh+1];
  // B tile: K x 16 classes   (lane = N col); zero-pad N>=3 branchlessly
  const int  nc  = sl < 3 ? sl : 0;
  const float mk = sl < 3 ? 1.f : 0.f;
  v2f b0; b0.x = fc_w[nc*8 + kh]*mk;     b0.y = fc_w[nc*8 + kh + 1]*mk;
  v2f b1; b1.x = fc_w[nc*8 + 4 + kh]*mk; b1.y = fc_w[nc*8 + 4 + kh + 1]*mk;
  v8f acc = {0.f,0.f,0.f,0.f,0.f,0.f,0.f,0.f};
  acc = __builtin_amdgcn_wmma_f32_16x16x4_f32(false, a0, false, b0, (short)0, acc, false, false);
  acc = __builtin_amdgcn_wmma_f32_16x16x4_f32(false, a1, false, b1, (short)0, acc, false, false);
  // D layout: vgpr v -> M = v + 8*half, lane -> N
  if (sl < 3){
    const float bias = fc_b[sl];
    const int mbase = tile*16 + half*8;
    #pragma unroll
    for (int v=0; v<8; v++)
      out[(mbase + v)*3 + sl] = acc[v] + bias;
  }
}

// ---------------- launch ----------------
extern "C" void kernel_launch(void* const* d_in, const int* in_sizes, int n_in,
                              void* d_out, int out_size, void* d_ws, size_t ws_size,
                              hipStream_t stream) {
  const float* X    = (const float*)d_in[0];
  const float* w1   = (const float*)d_in[1];
  const float* w2   = (const float*)d_in[2];
  const float* fc_w = (const float*)d_in[3];
  const float* fc_b = (const float*)d_in[4];
  float* out = (float*)d_out;
  float* g   = (float*)d_ws;                 // bs*8 floats of scratch

  const int bs = in_sizes[0] / 196;          // 8192
  const int blocks1 = (bs + 15) / 16;        // 16 samples per block
  qnn_main<<<blocks1, 256, 0, stream>>>(X, w1, w2, g, bs);

  const int tiles  = (bs + 15) / 16;         // 16 samples per wave
  const int blocks2 = (tiles + 7) / 8;       // 8 waves per block
  fc_wmma_kernel<<<blocks2, 256, 0, stream>>>(g, fc_w, fc_b, out, bs);
}